// MSAColumnGlobalAttention_89850715833162
// MI455X (gfx1250) — compile-verified
//
#include <hip/hip_runtime.h>

typedef float v2f __attribute__((ext_vector_type(2)));
typedef float v8f __attribute__((ext_vector_type(8)));

#define N_SEQ 2048
#define N_RES 384
#define C_IN  64
#define HC    64
#define SPAD  68          // LDS row stride (floats): 4-bank step -> conflict-free frag gathers
#define LN_EPS  1e-5f
#define EPS_ATT 1e-10f
#define INF_F   1.0e9f

static __device__ __forceinline__ v8f wmma_f32_16x16x4(v2f a, v2f b, v8f c) {
  // D = A(16x4,f32) * B(4x16,f32) + C(16x16,f32)
  return __builtin_amdgcn_wmma_f32_16x16x4_f32(false, a, false, b, (short)0, c, false, false);
}

// ---------------------------------------------------------------------------
// Kernel 0: zero pool accumulators; transpose wg/wo -> w[n][k] for B-fragments
// ---------------------------------------------------------------------------
__global__ __launch_bounds__(256) void prep_k(const float* __restrict__ wg,
                                              const float* __restrict__ wo,
                                              float* __restrict__ sumx,
                                              float* __restrict__ summask,
                                              float* __restrict__ wgT,
                                              float* __restrict__ woT) {
  int t = threadIdx.x;
  for (int i = t; i < N_RES * C_IN; i += 256) sumx[i] = 0.0f;
  for (int i = t; i < N_RES; i += 256) summask[i] = 0.0f;
  for (int i = t; i < 64 * 64; i += 256) {
    int k = i >> 6, n = i & 63;
    wgT[n * 64 + k] = wg[i];  // wg is [k][n] row-major -> wgT[n][k]
    woT[n * 64 + k] = wo[i];
  }
}

// ---------------------------------------------------------------------------
// Kernel A: LayerNorm + masked pooling + (k|v) projection via f32 WMMA
// grid (N_SEQ/128, N_RES), 256 threads = 8 waves, one 16-row s-tile per wave
// ---------------------------------------------------------------------------
__global__ __launch_bounds__(256) void msa_ln_kv_pool_k(
    const float* __restrict__ m, const float* __restrict__ mask,
    const float* __restrict__ g_lnw, const float* __restrict__ g_lnb,
    const float* __restrict__ wk, const float* __restrict__ wv,
    float* __restrict__ sumx, float* __restrict__ summask,
    float* __restrict__ kv) {
  __shared__ float wkvT[16 * SPAD];     // [n][k], n<8:wk col, n>=8:wv col
  __shared__ float lnw[64], lnb[64], pacc[64];
  __shared__ float xbuf[128 * SPAD];

  const int t = threadIdx.x;
  const int lane = t & 31;
  const int wid = t >> 5;
  const int r = blockIdx.y;
  const int sBase = blockIdx.x * 128;

  for (int i = t; i < 16 * 64; i += 256) {
    int n = i >> 6, k = i & 63;
    wkvT[n * SPAD + k] = (n < 8) ? wk[k * 8 + n] : wv[k * 8 + (n - 8)];
  }
  if (t < 64) { lnw[t] = g_lnw[t]; lnb[t] = g_lnb[t]; pacc[t] = 0.0f; }
  __syncthreads();

  const int row16 = lane & 15;     // row within tile
  const int half  = lane >> 4;     // which 32-column half this lane owns
  const int srow  = sBase + wid * 16 + row16;

  float v[32];
  const float* mrow = m + ((size_t)srow * N_RES + r) * C_IN + half * 32;
  #pragma unroll
  for (int j = 0; j < 8; ++j) {
    float4 f = *(const float4*)(mrow + j * 4);
    v[j*4+0] = f.x; v[j*4+1] = f.y; v[j*4+2] = f.z; v[j*4+3] = f.w;
  }
  float s = 0.f;
  #pragma unroll
  for (int j = 0; j < 32; ++j) s += v[j];
  s += __shfl_xor(s, 16);
  float mu = s * (1.0f / 64.0f);
  float vs = 0.f;
  #pragma unroll
  for (int j = 0; j < 32; ++j) { float d = v[j] - mu; vs += d * d; }
  vs += __shfl_xor(vs, 16);
  float rstd = rsqrtf(vs * (1.0f / 64.0f) + LN_EPS);

  float msk = mask[(size_t)srow * N_RES + r];
  float* xr = &xbuf[(wid * 16 + row16) * SPAD + half * 32];
  #pragma unroll
  for (int j4 = 0; j4 < 8; ++j4) {
    float4 w4 = *(const float4*)&lnw[half * 32 + j4 * 4];
    float4 b4 = *(const float4*)&lnb[half * 32 + j4 * 4];
    float4 o;
    o.x = (v[j4*4+0] - mu) * rstd * w4.x + b4.x;
    o.y = (v[j4*4+1] - mu) * rstd * w4.y + b4.y;
    o.z = (v[j4*4+2] - mu) * rstd * w4.z + b4.z;
    o.w = (v[j4*4+3] - mu) * rstd * w4.w + b4.w;
    *(float4*)(xr + j4 * 4) = o;
    atomicAdd(&pacc[half * 32 + j4 * 4 + 0], o.x * msk);
    atomicAdd(&pacc[half * 32 + j4 * 4 + 1], o.y * msk);
    atomicAdd(&pacc[half * 32 + j4 * 4 + 2], o.z * msk);
    atomicAdd(&pacc[half * 32 + j4 * 4 + 3], o.w * msk);
  }
  // mask sum over the 16 rows of this tile (halves hold duplicate masks)
  float msum = msk;
  msum += __shfl_xor(msum, 1);
  msum += __shfl_xor(msum, 2);
  msum += __shfl_xor(msum, 4);
  msum += __shfl_xor(msum, 8);
  if (lane == 0) atomicAdd(&summask[r], msum);
  __syncthreads();
  if (t < 64) atomicAdd(&sumx[r * 64 + t], pacc[t]);

  // kv = x(16x64) @ wkv(64x16) via 16 chained f32 WMMAs (K=4 each)
  v8f acc = {};
  const float* xw = &xbuf[(wid * 16) * SPAD];
  #pragma unroll
  for (int kc = 0; kc < 16; ++kc) {
    int k0 = kc * 4 + 2 * half;
    v2f a = *(const v2f*)(xw + row16 * SPAD + k0);   // A: M=row16, K=k0..k0+1
    v2f b = *(const v2f*)(&wkvT[row16 * SPAD + k0]); // B: N=row16, K=k0..k0+1
    acc = wmma_f32_16x16x4(a, b, acc);
  }
  #pragma unroll
  for (int j = 0; j < 8; ++j) {                      // D: M=j+8*half, N=row16
    int sr = sBase + wid * 16 + j + 8 * half;
    kv[((size_t)r * N_SEQ + sr) * 16 + row16] = acc[j];
  }
}

// ---------------------------------------------------------------------------
// Kernel B: per-r pooled query + two-pass softmax attention over S (L2-resident kv)
// grid N_RES, 256 threads
// ---------------------------------------------------------------------------
__global__ __launch_bounds__(256) void msa_attn_k(
    const float* __restrict__ mask, const float* __restrict__ wq,
    const float* __restrict__ sumx, const float* __restrict__ summask,
    const float* __restrict__ kv, float* __restrict__ og) {
  __shared__ float qs[64];
  __shared__ float wredm[8][8];
  __shared__ float wredo[8][72];
  const int t = threadIdx.x, lane = t & 31, wid = t >> 5;
  const int r = blockIdx.x;
  const float* kvr = kv + (size_t)r * N_SEQ * 16;

  if (t < 64) {
    float inv = 1.0f / (summask[r] + EPS_ATT);
    float q = 0.f;
    for (int c = 0; c < 64; ++c) q += sumx[r * 64 + c] * wq[c * 64 + t];
    qs[t] = q * inv * 0.35355339059327373f;  // * Ch^-0.5
  }
  __syncthreads();

  float qreg[64];
  #pragma unroll
  for (int j = 0; j < 64; ++j) qreg[j] = qs[j];

  float mh[8];
  #pragma unroll
  for (int h = 0; h < 8; ++h) mh[h] = -3.0e38f;

  for (int i = 0; i < N_SEQ / 256; ++i) {
    int sidx = t + i * 256;
    float bias = INF_F * (mask[(size_t)sidx * N_RES + r] - 1.0f);
    float4 k0 = *(const float4*)(kvr + sidx * 16);
    float4 k1 = *(const float4*)(kvr + sidx * 16 + 4);
    #pragma unroll
    for (int h = 0; h < 8; ++h) {
      float l = bias;
      l += qreg[h*8+0]*k0.x + qreg[h*8+1]*k0.y + qreg[h*8+2]*k0.z + qreg[h*8+3]*k0.w;
      l += qreg[h*8+4]*k1.x + qreg[h*8+5]*k1.y + qreg[h*8+6]*k1.z + qreg[h*8+7]*k1.w;
      mh[h] = fmaxf(mh[h], l);
    }
  }
  #pragma unroll
  for (int off = 16; off > 0; off >>= 1)
    #pragma unroll
    for (int h = 0; h < 8; ++h) mh[h] = fmaxf(mh[h], __shfl_xor(mh[h], off));
  if (lane == 0) {
    #pragma unroll
    for (int h = 0; h < 8; ++h) wredm[wid][h] = mh[h];
  }
  __syncthreads();
  float gm[8];
  #pragma unroll
  for (int h = 0; h < 8; ++h) {
    float x = wredm[0][h];
    #pragma unroll
    for (int w2 = 1; w2 < 8; ++w2) x = fmaxf(x, wredm[w2][h]);
    gm[h] = x;
  }

  float sh[8];
  float oacc[64];
  #pragma unroll
  for (int h = 0; h < 8; ++h) sh[h] = 0.f;
  #pragma unroll
  for (int j = 0; j < 64; ++j) oacc[j] = 0.f;

  for (int i = 0; i < N_SEQ / 256; ++i) {
    int sidx = t + i * 256;
    float bias = INF_F * (mask[(size_t)sidx * N_RES + r] - 1.0f);
    float4 k0 = *(const float4*)(kvr + sidx * 16);
    float4 k1 = *(const float4*)(kvr + sidx * 16 + 4);
    float4 v0 = *(const float4*)(kvr + sidx * 16 + 8);
    float4 v1 = *(const float4*)(kvr + sidx * 16 + 12);
    #pragma unroll
    for (int h = 0; h < 8; ++h) {
      float l = bias;
      l += qreg[h*8+0]*k0.x + qreg[h*8+1]*k0.y + qreg[h*8+2]*k0.z + qreg[h*8+3]*k0.w;
      l += qreg[h*8+4]*k1.x + qreg[h*8+5]*k1.y + qreg[h*8+6]*k1.z + qreg[h*8+7]*k1.w;
      float p = __expf(l - gm[h]);
      sh[h] += p;
      oacc[h*8+0] += p * v0.x; oacc[h*8+1] += p * v0.y;
      oacc[h*8+2] += p * v0.z; oacc[h*8+3] += p * v0.w;
      oacc[h*8+4] += p * v1.x; oacc[h*8+5] += p * v1.y;
      oacc[h*8+6] += p * v1.z; oacc[h*8+7] += p * v1.w;
    }
  }
  #pragma unroll
  for (int off = 16; off > 0; off >>= 1) {
    #pragma unroll
    for (int h = 0; h < 8; ++h) sh[h] += __shfl_xor(sh[h], off);
    #pragma unroll
    for (int j = 0; j < 64; ++j) oacc[j] += __shfl_xor(oacc[j], off);
  }
  if (lane == 0) {
    #pragma unroll
    for (int h = 0; h < 8; ++h) wredo[wid][h] = sh[h];
    #pragma unroll
    for (int j = 0; j < 64; ++j) wredo[wid][8 + j] = oacc[j];
  }
  __syncthreads();
  if (t < 64) {
    int h = t >> 3;
    float den = 0.f, num = 0.f;
    #pragma unroll
    for (int w2 = 0; w2 < 8; ++w2) { den += wredo[w2][h]; num += wredo[w2][8 + t]; }
    og[r * 64 + t] = num / den;
  }
}

// ---------------------------------------------------------------------------
// Kernel C: recompute LN, GEMM x@wg (+bg) -> sigmoid gate * o, GEMM @wo + bo
// grid (N_SEQ/128, N_RES), 256 threads = 8 waves, 16-row tile per wave
// ---------------------------------------------------------------------------
__global__ __launch_bounds__(256) void msa_gate_out_k(
    const float* __restrict__ m,
    const float* __restrict__ g_lnw, const float* __restrict__ g_lnb,
    const float* __restrict__ g_bg, const float* __restrict__ g_bo,
    const float* __restrict__ wgT_g, const float* __restrict__ woT_g,
    const float* __restrict__ og, float* __restrict__ out) {
  extern __shared__ float smem[];
  float* wgT  = smem;                   // 64*SPAD
  float* woT  = wgT + 64 * SPAD;        // 64*SPAD
  float* xbuf = woT + 64 * SPAD;        // 128*SPAD (x, then reused for gated tmp)
  float* lnw  = xbuf + 128 * SPAD;      // 64
  float* lnb  = lnw + 64;
  float* bgs  = lnb + 64;
  float* bos  = bgs + 64;
  float* ogs  = bos + 64;

  const int t = threadIdx.x, lane = t & 31, wid = t >> 5;
  const int r = blockIdx.y;
  const int sBase = blockIdx.x * 128;

  for (int i = t; i < 64 * 64; i += 256) {
    int n = i >> 6, k = i & 63;
    wgT[n * SPAD + k] = wgT_g[i];
    woT[n * SPAD + k] = woT_g[i];
  }
  if (t < 64) {
    lnw[t] = g_lnw[t]; lnb[t] = g_lnb[t];
    bgs[t] = g_bg[t];  bos[t] = g_bo[t];
    ogs[t] = og[r * 64 + t];
  }
  __syncthreads();

  const int row16 = lane & 15, half = lane >> 4;
  const int srow = sBase + wid * 16 + row16;

  float v[32];
  const float* mrow = m + ((size_t)srow * N_RES + r) * C_IN + half * 32;
  #pragma unroll
  for (int j = 0; j < 8; ++j) {
    float4 f = *(const float4*)(mrow + j * 4);
    v[j*4+0] = f.x; v[j*4+1] = f.y; v[j*4+2] = f.z; v[j*4+3] = f.w;
  }
  float s = 0.f;
  #pragma unroll
  for (int j = 0; j < 32; ++j) s += v[j];
  s += __shfl_xor(s, 16);
  float mu = s * (1.0f / 64.0f);
  float vs = 0.f;
  #pragma unroll
  for (int j = 0; j < 32; ++j) { float d = v[j] - mu; vs += d * d; }
  vs += __shfl_xor(vs, 16);
  float rstd = rsqrtf(vs * (1.0f / 64.0f) + LN_EPS);

  float* xr = &xbuf[(wid * 16 + row16) * SPAD + half * 32];
  #pragma unroll
  for (int j4 = 0; j4 < 8; ++j4) {
    float4 w4 = *(const float4*)&lnw[half * 32 + j4 * 4];
    float4 b4 = *(const float4*)&lnb[half * 32 + j4 * 4];
    float4 o;
    o.x = (v[j4*4+0] - mu) * rstd * w4.x + b4.x;
    o.y = (v[j4*4+1] - mu) * rstd * w4.y + b4.y;
    o.z = (v[j4*4+2] - mu) * rstd * w4.z + b4.z;
    o.w = (v[j4*4+3] - mu) * rstd * w4.w + b4.w;
    *(float4*)(xr + j4 * 4) = o;
  }
  __syncthreads();   // cross-lane: fragment gathers read other lanes' x

  // A-fragments of x, kept in registers across all 4 N-tiles
  v2f af[16];
  const float* xw = &xbuf[(wid * 16 + row16) * SPAD + 2 * half];
  #pragma unroll
  for (int kc = 0; kc < 16; ++kc) af[kc] = *(const v2f*)(xw + kc * 4);

  // GEMM1: G = x @ wg; gated tmp = sigmoid(G + bg) * o  (written back into xbuf)
  #pragma unroll
  for (int nt = 0; nt < 4; ++nt) {
    v8f acc = {};
    const float* bp = &wgT[(nt * 16 + row16) * SPAD + 2 * half];
    #pragma unroll
    for (int kc = 0; kc < 16; ++kc) {
      v2f b = *(const v2f*)(bp + kc * 4);
      acc = wmma_f32_16x16x4(af[kc], b, acc);
    }
    int col = nt * 16 + row16;
    float gb = bgs[col], oo = ogs[col];
    #pragma unroll
    for (int j = 0; j < 8; ++j) {
      float sg = 1.0f / (1.0f + __expf(-(acc[j] + gb)));
      xbuf[(wid * 16 + j + 8 * half) * SPAD + col] = sg * oo;
    }
  }
  __syncthreads();   // cross-lane: tmp fragment gathers read other lanes' stores

  v2f tf[16];
  const float* tw = &xbuf[(wid * 16 + row16) * SPAD + 2 * half];
  #pragma unroll
  for (int kc = 0; kc < 16; ++kc) tf[kc] = *(const v2f*)(tw + kc * 4);

  // GEMM2: out = tmp @ wo + bo
  #pragma unroll
  for (int nt = 0; nt < 4; ++nt) {
    v8f acc = {};
    const float* bp = &woT[(nt * 16 + row16) * SPAD + 2 * half];
    #pragma unroll
    for (int kc = 0; kc < 16; ++kc) {
      v2f b = *(const v2f*)(bp + kc * 4);
      acc = wmma_f32_16x16x4(tf[kc], b, acc);
    }
    int col = nt * 16 + row16;
    float bb = bos[col];
    #pragma unroll
    for (int j = 0; j < 8; ++j) {
      int sr = sBase + wid * 16 + j + 8 * half;
      out[((size_t)sr * N_RES + r) * C_IN + col] = acc[j] + bb;
    }
  }
}

// ---------------------------------------------------------------------------
extern "C" void kernel_launch(void* const* d_in, const int* in_sizes, int n_in,
                              void* d_out, int out_size, void* d_ws, size_t ws_size,
                              hipStream_t stream) {
  const float* m    = (const float*)d_in[0];
  const float* mask = (const float*)d_in[1];
  const float* ln_w = (const float*)d_in[2];
  const float* ln_b = (const float*)d_in[3];
  const float* wq   = (const float*)d_in[4];
  const float* wk   = (const float*)d_in[5];
  const float* wv   = (const float*)d_in[6];
  const float* wg   = (const float*)d_in[7];
  const float* bg   = (const float*)d_in[8];
  const float* wo   = (const float*)d_in[9];
  const float* bo   = (const float*)d_in[10];
  float* out = (float*)d_out;

  float* w       = (float*)d_ws;
  float* sumx    = w;                          // N_RES*64
  float* summask = sumx + N_RES * C_IN;        // N_RES
  float* wgT     = summask + N_RES;            // 64*64
  float* woT     = wgT + 64 * 64;              // 64*64
  float* og      = woT + 64 * 64;              // N_RES*64
  float* kv      = og + N_RES * HC;            // N_RES*N_SEQ*16 (~50 MB)

  prep_k<<<1, 256, 0, stream>>>(wg, wo, sumx, summask, wgT, woT);
  msa_ln_kv_pool_k<<<dim3(N_SEQ / 128, N_RES), 256, 0, stream>>>(
      m, mask, ln_w, ln_b, wk, wv, sumx, summask, kv);
  msa_attn_k<<<N_RES, 256, 0, stream>>>(mask, wq, sumx, summask, kv, og);
  size_t smemC = (size_t)(64 * SPAD * 2 + 128 * SPAD + 5 * 64) * sizeof(float);
  msa_gate_out_k<<<dim3(N_SEQ / 128, N_RES), 256, smemC, stream>>>(
      m, ln_w, ln_b, bg, bo, wgT, woT, og, out);
}